// Net_68839735820425
// MI455X (gfx1250) — compile-verified
//
#include <hip/hip_runtime.h>
#include <hip/hip_bf16.h>

#define N_NODES 12288
#define N_FEAT  500
#define N_CLASSES 16
#define N_EDGES (N_NODES * 32)
#define WPR (N_NODES / 32)   /* 384 u32 words per adjacency row */

typedef __attribute__((ext_vector_type(2))) float v2f;
typedef __attribute__((ext_vector_type(8))) float v8f;

// ---------------------------------------------------------------------------
// Kernel 1: h0 = softmax(x @ W, axis=1).  16 lanes per node (one class each).
// x row is broadcast across the 16 lanes; W (32 KB) is L2/L0 resident.
// ---------------------------------------------------------------------------
__global__ __launch_bounds__(256) void head_softmax_kernel(
    const float* __restrict__ x, const float* __restrict__ W,
    float* __restrict__ h0) {
  int t = blockIdx.x * blockDim.x + threadIdx.x;
  int node = t >> 4;
  int c = t & 15;
  if (node >= N_NODES) return;

  const float* xr = x + (long long)node * N_FEAT;
  float acc = 0.0f;
#pragma unroll 4
  for (int j = 0; j < N_FEAT; ++j)
    acc = fmaf(xr[j], W[j * N_CLASSES + c], acc);

  // softmax across the 16 lanes of this node (xor masks 1..8 stay in-group)
  float m = acc;
#pragma unroll
  for (int off = 8; off >= 1; off >>= 1)
    m = fmaxf(m, __shfl_xor(m, off, 32));
  float e = __expf(acc - m);
  float s = e;
#pragma unroll
  for (int off = 8; off >= 1; off >>= 1)
    s += __shfl_xor(s, off, 32);

  h0[node * N_CLASSES + c] = e / s;
}

// ---------------------------------------------------------------------------
// Kernel 2: scatter edges into a dense bit-matrix (atomicOr dedups exactly
// like jnp .at[].set(1) on a dense adjacency).
// ---------------------------------------------------------------------------
__global__ __launch_bounds__(256) void scatter_edges_kernel(
    const int* __restrict__ src, const int* __restrict__ dst,
    unsigned int* __restrict__ adj) {
  int e = blockIdx.x * blockDim.x + threadIdx.x;
  if (e >= N_EDGES) return;
  int r = src[e];
  int c = dst[e];
  atomicOr(adj + (long long)r * WPR + (c >> 5), 1u << (c & 31));
}

// ---------------------------------------------------------------------------
// Kernel 3: deg[i] = popcount(row i) + 1 (the +eye term); dis = deg^-1/2.
// Note: if a self-edge exists, A_ii = 2 in the reference; the bit is counted
// by popcount and the eye contributes the extra +1 — exact match.
// ---------------------------------------------------------------------------
__global__ __launch_bounds__(256) void degree_kernel(
    const unsigned int* __restrict__ adj, float* __restrict__ dis) {
  int i = blockIdx.x * blockDim.x + threadIdx.x;
  if (i >= N_NODES) return;
  const unsigned int* row = adj + (long long)i * WPR;
  int cnt = 1;  // self loop from +eye
#pragma unroll 4
  for (int w = 0; w < WPR; ++w) cnt += __popc(row[w]);
  dis[i] = rsqrtf((float)cnt);
}

// ---------------------------------------------------------------------------
// Kernel 4: log(x_finally) = log( D^-1/2 (A + I) D^-1/2 @ h0 ), sparse.
// 16 lanes per row, one class each; bit-scan gather over the row's words.
// ---------------------------------------------------------------------------
__global__ __launch_bounds__(256) void spmm_log_kernel(
    const unsigned int* __restrict__ adj, const float* __restrict__ dis,
    const float* __restrict__ h0, float* __restrict__ out) {
  int t = blockIdx.x * blockDim.x + threadIdx.x;
  int i = t >> 4;
  int c = t & 15;
  if (i >= N_NODES) return;

  const unsigned int* row = adj + (long long)i * WPR;
  float di = dis[i];
  // eye contribution (the +1 on the diagonal beyond any self-edge bit)
  float acc = di * h0[i * N_CLASSES + c];
  for (int w = 0; w < WPR; ++w) {
    unsigned int bits = row[w];
    while (bits) {
      int b = __ffs(bits) - 1;
      bits &= bits - 1u;
      int j = (w << 5) + b;
      acc = fmaf(dis[j], h0[j * N_CLASSES + c], acc);
    }
  }
  out[i * N_CLASSES + c] = __logf(di * acc);
}

// ---------------------------------------------------------------------------
// Kernel 5: sim = h0 @ h0^T via V_WMMA_F32_16X16X4_F32 (wave32).
// One 16x16 output tile per wave, K=16 in 4 WMMA steps.  Store-bound:
// 604 MB of output, h0 (786 KB) stays in L2.  EXEC is all-1s: the grid
// covers tiles exactly and there is no divergence.
//
// fp32 A 16x4 layout (ISA 7.12.2): lane l holds M=l%16; VGPR v holds
// K = kbase + 2*(l/16) + v  -> one float2 load per lane per step.
// B 4x16: lane l holds N=l%16 with the same K striping; B = h0^T so the
// per-lane source row is the *column* node.
// C/D: VGPR r, lane l -> (M = r + 8*(l/16), N = l%16).
// ---------------------------------------------------------------------------
__global__ __launch_bounds__(256) void sim_wmma_kernel(
    const float* __restrict__ h0, float* __restrict__ sim) {
  const int lane = threadIdx.x & 31;
  const int wave = threadIdx.x >> 5;
  const int tilesPerRow = N_NODES / 16;  // 768
  const long long tileId = (long long)blockIdx.x * 8 + wave;
  const int tm = (int)(tileId / tilesPerRow);
  const int tn = (int)(tileId % tilesPerRow);

  const int half = lane >> 4;   // 0 or 1
  const int l15  = lane & 15;
  const float* aRow = h0 + (long long)(tm * 16 + l15) * N_CLASSES;
  const float* bRow = h0 + (long long)(tn * 16 + l15) * N_CLASSES;

  v8f c = {0.f, 0.f, 0.f, 0.f, 0.f, 0.f, 0.f, 0.f};
#pragma unroll
  for (int s = 0; s < 4; ++s) {
    const int k0 = s * 4 + 2 * half;
    v2f a = *(const v2f*)(aRow + k0);
    v2f b = *(const v2f*)(bRow + k0);
    c = __builtin_amdgcn_wmma_f32_16x16x4_f32(
        /*neg_a=*/false, a, /*neg_b=*/false, b,
        /*c_mod=*/(short)0, c, /*reuse_a=*/false, /*reuse_b=*/false);
  }

  float* base = sim + (long long)(tm * 16) * N_NODES + tn * 16;
#pragma unroll
  for (int r = 0; r < 8; ++r) {
    int m = r + 8 * half;
    base[(long long)m * N_NODES + l15] = c[r];
  }
}

// ---------------------------------------------------------------------------
extern "C" void kernel_launch(void* const* d_in, const int* in_sizes, int n_in,
                              void* d_out, int out_size, void* d_ws, size_t ws_size,
                              hipStream_t stream) {
  const float* x  = (const float*)d_in[0];
  const int* eidx = (const int*)d_in[1];           // [2, E]: row0=src, row1=dst
  const float* W  = (const float*)d_in[2];

  const int* src = eidx;
  const int* dst = eidx + N_EDGES;

  // workspace layout
  unsigned char* ws = (unsigned char*)d_ws;
  unsigned int* adj = (unsigned int*)ws;                               // 18.87 MB
  float* h0  = (float*)(ws + (size_t)N_NODES * WPR * sizeof(unsigned int));
  float* dis = (float*)((unsigned char*)h0 + (size_t)N_NODES * N_CLASSES * sizeof(float));

  float* simOut = (float*)d_out;                                       // [N, N]
  float* logOut = simOut + (long long)N_NODES * N_NODES;               // [N, C]

  // 0) clear adjacency bits (memset node is graph-capture safe)
  hipMemsetAsync(adj, 0, (size_t)N_NODES * WPR * sizeof(unsigned int), stream);

  // 1) h0 = softmax(x @ W)
  head_softmax_kernel<<<(N_NODES * N_CLASSES) / 256, 256, 0, stream>>>(x, W, h0);

  // 2) adjacency bit-matrix
  scatter_edges_kernel<<<N_EDGES / 256, 256, 0, stream>>>(src, dst, adj);

  // 3) degrees -> D^{-1/2}
  degree_kernel<<<N_NODES / 256, 256, 0, stream>>>(adj, dis);

  // 4) log(adj_t @ h0)
  spmm_log_kernel<<<(N_NODES * N_CLASSES) / 256, 256, 0, stream>>>(adj, dis, h0, logOut);

  // 5) sim = h0 @ h0^T  (WMMA f32, 768x768 tiles, 8 waves/block)
  const int tiles = (N_NODES / 16) * (N_NODES / 16);   // 589824
  sim_wmma_kernel<<<tiles / 8, 256, 0, stream>>>(h0, simOut);
}